// URPE_36807869726820
// MI455X (gfx1250) — compile-verified
//
#include <hip/hip_runtime.h>
#include <hip/hip_bf16.h>
#include <cstdint>

#define B_ 2
#define H_ 16
#define L_ 2048
#define NVALS (2 * L_ - 1)   // 4095
#define ROWS 16
#define NTHREADS 256

// --- CDNA5 async global->LDS path (guarded; host pass / old toolchains fall back) ---
#if defined(__has_builtin)
#  if __has_builtin(__builtin_amdgcn_global_load_async_to_lds_b128) && \
      __has_builtin(__builtin_amdgcn_s_wait_asynccnt)
#    define USE_ASYNC_LDS 1
#  endif
#endif
#ifndef USE_ASYNC_LDS
#  define USE_ASYNC_LDS 0
#endif

typedef int v4i __attribute__((vector_size(16)));
typedef __attribute__((address_space(1))) v4i GV4I;   // global (AS1) 16-byte vector
typedef __attribute__((address_space(3))) v4i LV4I;   // LDS    (AS3) 16-byte vector

// toe[i][j] = vals[(j-i) mod (2L-1)] with
// vals[k] = wh[L+k]        for k in [0, L)      (the "r" half)
// vals[k] = wh[2L-1-k]     for k in [L, 2L-2]   (reversed c[1:])
// smem holds the raw per-head weight row wh[0..2L).
__device__ __forceinline__ float toe_val(const float* __restrict__ smem, int i, int j) {
  int idx = j - i;
  idx += (idx < 0) ? NVALS : 0;                       // [0, 4094]
  int widx = (idx < L_) ? (L_ + idx) : (NVALS - idx); // index into raw weight row
  return smem[widx];
}

__global__ __launch_bounds__(NTHREADS) void urpe_mul_kernel(
    const float* __restrict__ probs, const float* __restrict__ w,
    float* __restrict__ out) {
  __shared__ float svals[2 * L_];  // 16 KiB: raw weight row for this head

  const int tid = threadIdx.x;
  const int h = blockIdx.y;
  const int b = blockIdx.z;
  const float* wh = w + (size_t)h * (2 * L_);

  // Stage the 16 KiB weight row into LDS.
#if USE_ASYNC_LDS
  {
#pragma unroll
    for (int it = 0; it < (2 * L_) / (NTHREADS * 4); ++it) {  // 4 iterations of b128
      const int f = (it * NTHREADS + tid) * 4;
      GV4I* g = (GV4I*)(uintptr_t)(wh + f);
      // Low 32 bits of a generic LDS pointer are the LDS byte offset (ISA 10.2).
      LV4I* l = (LV4I*)(uint32_t)(uintptr_t)(&svals[f]);
      __builtin_amdgcn_global_load_async_to_lds_b128(g, l, 0, 0);
    }
    __builtin_amdgcn_s_wait_asynccnt(0);
  }
#else
  for (int f = tid * 4; f < 2 * L_; f += NTHREADS * 4) {
    *(float4*)(&svals[f]) = *(const float4*)(wh + f);
  }
#endif
  __syncthreads();

  const size_t plane = (((size_t)b * H_) + h) * (size_t)L_ * (size_t)L_;
  const int row0 = blockIdx.x * ROWS;

  for (int r = 0; r < ROWS; ++r) {
    const int i = row0 + r;
    const float4* __restrict__ prow = (const float4*)(probs + plane + (size_t)i * L_);
    float4* __restrict__ orow       = (float4*)(out   + plane + (size_t)i * L_);

#pragma unroll
    for (int half = 0; half < 2; ++half) {
      const int t = half * NTHREADS + tid;  // float4 index within the row
      const int j0 = t * 4;
      float4 p = prow[t];
      float4 o;
      o.x = p.x * toe_val(svals, i, j0 + 0);
      o.y = p.y * toe_val(svals, i, j0 + 1);
      o.z = p.z * toe_val(svals, i, j0 + 2);
      o.w = p.w * toe_val(svals, i, j0 + 3);
      orow[t] = o;
    }

    // Speculative prefetch of the next row (8 KiB; 256 lanes x 32B covers it).
    if (r + 1 < ROWS) {
      const float* np = probs + plane + (size_t)(i + 1) * L_ + (size_t)tid * 8;
      __builtin_prefetch(np, 0, 0);
    }
  }
}

extern "C" void kernel_launch(void* const* d_in, const int* in_sizes, int n_in,
                              void* d_out, int out_size, void* d_ws, size_t ws_size,
                              hipStream_t stream) {
  const float* probs = (const float*)d_in[0];   // (B,H,L,L) f32
  const float* w     = (const float*)d_in[1];   // (H,2L)    f32
  float* out         = (float*)d_out;           // (B,H,L,L) f32

  dim3 grid(L_ / ROWS, H_, B_);   // 128 x 16 x 2 = 4096 blocks
  urpe_mul_kernel<<<grid, NTHREADS, 0, stream>>>(probs, w, out);

  (void)in_sizes; (void)n_in; (void)out_size; (void)d_ws; (void)ws_size;
}